// GCN_1_3246995276079
// MI455X (gfx1250) — compile-verified
//
#include <hip/hip_runtime.h>

typedef __attribute__((ext_vector_type(2))) float v2f;
typedef __attribute__((ext_vector_type(8))) float v8f;

#define D 64  // D_IN == D_HID == D_OUT == 64

// ---------------- helpers ----------------

__global__ void zero_f32(float* __restrict__ p, int n) {
  int i = blockIdx.x * blockDim.x + threadIdx.x;
  if (i < n) p[i] = 0.0f;
}

// deg[d] += 1 for each edge endpoint (self-loop added analytically later)
__global__ void deg_count(const int* __restrict__ dst, float* __restrict__ deg, int n_edges) {
  int e = blockIdx.x * blockDim.x + threadIdx.x;
  if (e < n_edges) atomicAdd(&deg[dst[e]], 1.0f);
}

// deg -> 1/sqrt(deg + 1)   (+1 = self loop; always > 0)
__global__ void invsqrt_finish(float* __restrict__ d, int n) {
  int i = blockIdx.x * blockDim.x + threadIdx.x;
  if (i < n) d[i] = 1.0f / sqrtf(d[i] + 1.0f);
}

__global__ void edge_norm(const int* __restrict__ src, const int* __restrict__ dst,
                          const float* __restrict__ inv, float* __restrict__ norm,
                          int n_edges) {
  int e = blockIdx.x * blockDim.x + threadIdx.x;
  if (e < n_edges) norm[e] = inv[src[e]] * inv[dst[e]];
}

// ---------------- WMMA GEMM: T = (RELU? max(A,0):A) @ W ----------------
// A: [n_rows x 64] row-major, W: [64 x 64] row-major, T: [n_rows x 64].
// One wave per 16x16 output tile; block = 4 waves = the 4 column tiles of a
// 16-row stripe. K looped in chunks of 4 -> 16x v_wmma_f32_16x16x4_f32.
// Requires n_rows % 16 == 0 (100000 = 6250*16) so EXEC is all-ones.
template <bool RELU>
__global__ void gemm16_wmma(const float* __restrict__ A, const float* __restrict__ W,
                            float* __restrict__ T) {
  const int lane = threadIdx.x & 31;
  const int wave = threadIdx.x >> 5;     // 0..3 -> column tile
  const int half = lane >> 4;            // lane half selects K pair / row half
  const int l    = lane & 15;
  const int row0 = blockIdx.x << 4;
  const int c0   = wave << 4;

  const float* __restrict__ arow = A + (size_t)(row0 + l) * D;

  v8f acc = {};
#pragma unroll
  for (int kk = 0; kk < D; kk += 4) {
    const int ka = kk + half * 2;        // lanes 0-15: K=kk,kk+1 ; lanes 16-31: K=kk+2,kk+3
    float a0 = arow[ka];
    float a1 = arow[ka + 1];
    if (RELU) { a0 = fmaxf(a0, 0.0f); a1 = fmaxf(a1, 0.0f); }
    v2f a = {a0, a1};
    v2f b = { W[ka * D + c0 + l], W[(ka + 1) * D + c0 + l] };
    acc = __builtin_amdgcn_wmma_f32_16x16x4_f32(
        /*neg_a=*/false, a, /*neg_b=*/false, b,
        /*c_mod=*/(short)0, acc, /*reuse_a=*/false, /*reuse_b=*/false);
  }

  // C/D layout: VGPR j -> (row0+j, c0+lane) for lanes 0-15, (row0+8+j, ...) for 16-31
  float* __restrict__ orow = T + (size_t)(row0 + half * 8) * D + c0 + l;
#pragma unroll
  for (int j = 0; j < 8; ++j) orow[(size_t)j * D] = acc[j];
}

// out[i,f] = T[i,f] * inv[i]^2 + bias[f]   (self-loop term + bias, no atomics)
__global__ void self_bias_init(const float* __restrict__ T, const float* __restrict__ inv,
                               const float* __restrict__ bias, float* __restrict__ out,
                               int total) {
  int i = blockIdx.x * blockDim.x + threadIdx.x;
  if (i >= total) return;
  float is = inv[i >> 6];
  out[i] = T[i] * is * is + bias[i & (D - 1)];
}

// out[dst] += T[src] * norm[e]  — 32 lanes per edge, float2 gather, fp32 atomics
__global__ void edge_aggregate(const int* __restrict__ src, const int* __restrict__ dst,
                               const float* __restrict__ norm, const float* __restrict__ T,
                               float* __restrict__ out, int n_edges) {
  int e = blockIdx.x * (blockDim.x >> 5) + (threadIdx.x >> 5);
  if (e >= n_edges) return;
  int f2 = (threadIdx.x & 31) * 2;
  int s = src[e];
  int d = dst[e];
  float nv = norm[e];
  const float2 t = *(const float2*)(T + (size_t)s * D + f2);
  float* o = out + (size_t)d * D + f2;
  atomicAdd(o,     t.x * nv);
  atomicAdd(o + 1, t.y * nv);
}

// ---------------- launch ----------------

static inline size_t align256(size_t x) { return (x + 255) & ~(size_t)255; }

extern "C" void kernel_launch(void* const* d_in, const int* in_sizes, int n_in,
                              void* d_out, int out_size, void* d_ws, size_t ws_size,
                              hipStream_t stream) {
  // inputs: V, E, X, W1, b1, W2, b2, W3, b3
  const int*   E  = (const int*)d_in[1];
  const float* X  = (const float*)d_in[2];
  const float* W1 = (const float*)d_in[3];
  const float* b1 = (const float*)d_in[4];
  const float* W2 = (const float*)d_in[5];
  const float* b2 = (const float*)d_in[6];
  const float* W3 = (const float*)d_in[7];
  const float* b3 = (const float*)d_in[8];
  float* out = (float*)d_out;

  const int n_edges = in_sizes[1] / 2;
  const int n_nodes = in_sizes[2] / D;
  const int* src = E;
  const int* dst = E + n_edges;

  // workspace layout
  char* ws = (char*)d_ws;
  float* inv  = (float*)ws; ws += align256((size_t)n_nodes * 4);
  float* norm = (float*)ws; ws += align256((size_t)n_edges * 4);
  float* T    = (float*)ws; ws += align256((size_t)n_nodes * D * 4);
  float* H    = (float*)ws;

  const int total = n_nodes * D;
  const int nb_nodes = (n_nodes + 255) / 256;
  const int nb_edges = (n_edges + 255) / 256;
  const int nb_total = (total + 255) / 256;
  const int nb_agg   = (n_edges + 7) / 8;       // 8 edges per 256-thread block
  const int nb_gemm  = n_nodes / 16;            // 100000/16 = 6250 exact

  // normalization (once, reused for all 3 layers)
  zero_f32<<<nb_nodes, 256, 0, stream>>>(inv, n_nodes);
  deg_count<<<nb_edges, 256, 0, stream>>>(dst, inv, n_edges);
  invsqrt_finish<<<nb_nodes, 256, 0, stream>>>(inv, n_nodes);
  edge_norm<<<nb_edges, 256, 0, stream>>>(src, dst, inv, norm, n_edges);

  // layer 1: H = agg(X @ W1) + b1   (relu fused into next layer's A-load)
  gemm16_wmma<false><<<nb_gemm, 128, 0, stream>>>(X, W1, T);
  self_bias_init<<<nb_total, 256, 0, stream>>>(T, inv, b1, H, total);
  edge_aggregate<<<nb_agg, 256, 0, stream>>>(src, dst, norm, T, H, n_edges);

  // layer 2: H = agg(relu(H) @ W2) + b2
  gemm16_wmma<true><<<nb_gemm, 128, 0, stream>>>(H, W2, T);
  self_bias_init<<<nb_total, 256, 0, stream>>>(T, inv, b2, H, total);
  edge_aggregate<<<nb_agg, 256, 0, stream>>>(src, dst, norm, T, H, n_edges);

  // layer 3: out = agg(relu(H) @ W3) + b3   (no relu on output)
  gemm16_wmma<true><<<nb_gemm, 128, 0, stream>>>(H, W3, T);
  self_bias_init<<<nb_total, 256, 0, stream>>>(T, inv, b3, out, total);
  edge_aggregate<<<nb_agg, 256, 0, stream>>>(src, dst, norm, T, out, n_edges);
}